// skeaTopoLoss_56538949485004
// MI455X (gfx1250) — compile-verified
//
#include <hip/hip_runtime.h>
#include <hip/hip_bf16.h>
#include <stdint.h>

// Problem geometry (fixed by the reference)
#define B_        16
#define H_        1024
#define W_        1024
#define WW        32                 // u32 words per row (1024 bits)
#define IMG_WORDS (H_ * WW)          // 32768 words  = 128 KB per image mask
#define IMG_PIX   (H_ * W_)          // 1048576
#define TOT_WORDS (B_ * IMG_WORDS)   // 524288
#define TOT_PIX   (B_ * IMG_PIX)     // 16777216
#define SK_ITERS  32

typedef float v2f __attribute__((ext_vector_type(2)));
typedef float v8f __attribute__((ext_vector_type(8)));

// ---------------------------------------------------------------------------
// Bit-packed 3x3 morphology helpers. Bit j of word c = pixel x = c*32 + j.
// Erosion: AND of 3 horizontal neighbors, OOB = 1 (reduce_window init +inf).
// Dilation: OR  of 3 horizontal neighbors, OOB = 0 (reduce_window init -inf).
// ---------------------------------------------------------------------------
__device__ __forceinline__ unsigned hE(const unsigned* b, int r, int c) {
  const unsigned m  = b[(r << 5) + c];
  const unsigned lw = (c > 0)      ? b[(r << 5) + c - 1] : 0xFFFFFFFFu;
  const unsigned rw = (c < WW - 1) ? b[(r << 5) + c + 1] : 0xFFFFFFFFu;
  return m & ((m << 1) | (lw >> 31)) & ((m >> 1) | (rw << 31));
}
__device__ __forceinline__ unsigned hD(const unsigned* b, int r, int c) {
  const unsigned m  = b[(r << 5) + c];
  const unsigned lw = (c > 0)      ? b[(r << 5) + c - 1] : 0u;
  const unsigned rw = (c < WW - 1) ? b[(r << 5) + c + 1] : 0u;
  return m | (m << 1) | (lw >> 31) | (m >> 1) | (rw << 31);
}

// ---------------------------------------------------------------------------
// Kernel 0: zero the per-image counters (ws is poisoned; must re-init per call)
// ---------------------------------------------------------------------------
__global__ void init_kernel(unsigned* __restrict__ cnts) {
  if (threadIdx.x < B_) cnts[threadIdx.x] = 0u;
}

// ---------------------------------------------------------------------------
// Kernel 1: bit-pack target via wave32 ballot + per-image popcount
// block = 256 threads (8 waves, 8 words); 4096 blocks per image.
// ---------------------------------------------------------------------------
__global__ __launch_bounds__(256) void pack_kernel(const int* __restrict__ tgt,
                                                   unsigned* __restrict__ amask,
                                                   unsigned* __restrict__ cnts) {
  __shared__ unsigned bcnt;
  const int tid = threadIdx.x;
  if (tid == 0) bcnt = 0u;
  __syncthreads();
  const int p = blockIdx.x * 256 + tid;              // pixel index, lane = bit
  const int v = tgt[p];
  const unsigned bal = __builtin_amdgcn_ballot_w32(v != 0);
  if ((tid & 31) == 0) {
    amask[p >> 5] = bal;
    atomicAdd(&bcnt, (unsigned)__popc(bal));
  }
  __syncthreads();
  if (tid == 0) atomicAdd(&cnts[blockIdx.x >> 12], bcnt);   // 4096 blocks/image
}

// ---------------------------------------------------------------------------
// Kernel 2: skeletonize one full image in LDS (256 KB ping-pong of bitmasks).
// grid = (16 images, 2 variants: fore = a, back = ~a), block = 1024 threads.
// Wave w owns rows [w*32, w*32+32), lane = word-column -> conflict-free LDS.
// s accumulates in global (L2-resident).
// ---------------------------------------------------------------------------
__global__ __launch_bounds__(1024, 1) void sk_kernel(const unsigned* __restrict__ amask,
                                                     unsigned* __restrict__ fsk,
                                                     unsigned* __restrict__ bsk) {
  __shared__ unsigned sA[IMG_WORDS];
  __shared__ unsigned sB[IMG_WORDS];
  const int img     = blockIdx.x;
  const int variant = blockIdx.y;
  const int tid     = threadIdx.x;
  const unsigned* gm = amask + img * IMG_WORDS;
  unsigned* gs = (variant ? bsk : fsk) + img * IMG_WORDS;

  // Stage the packed mask into LDS with CDNA5 async global->LDS loads.
  {
    const unsigned base = (unsigned)tid * 32u;       // 32 words / thread, b128 x8
#pragma unroll
    for (int i = 0; i < 8; ++i) {
      const unsigned ldsb = (unsigned)(uintptr_t)(&sA[base + (unsigned)i * 4u]);
      const uint64_t ga   = (uint64_t)(uintptr_t)(gm + base + (unsigned)i * 4u);
      asm volatile("global_load_async_to_lds_b128 %0, %1, off"
                   :: "v"(ldsb), "v"(ga) : "memory");
    }
    asm volatile("s_wait_asynccnt 0" ::: "memory");
  }
  __syncthreads();
  if (variant) {                                     // back skeleton uses 1 - a
#pragma unroll 4
    for (int k = 0; k < 32; ++k) { const int w = tid + (k << 10); sA[w] = ~sA[w]; }
    __syncthreads();
  }

  const int c  = tid & 31;          // word-column
  const int r0 = (tid >> 5) << 5;   // first row owned by this wave

  for (int it = 0; it < SK_ITERS; ++it) {
    unsigned* cur = (it & 1) ? sB : sA;
    unsigned* nxt = (it & 1) ? sA : sB;
    // Pass A: nxt = erode(cur), rolling 3-row window of horizontal erosions.
    {
      unsigned hp = (r0 == 0) ? 0xFFFFFFFFu : hE(cur, r0 - 1, c);
      unsigned hc = hE(cur, r0, c);
#pragma unroll 4
      for (int k = 0; k < 32; ++k) {
        const int r = r0 + k;
        const unsigned hn = (r == H_ - 1) ? 0xFFFFFFFFu : hE(cur, r + 1, c);
        nxt[(r << 5) + c] = hp & hc & hn;
        hp = hc; hc = hn;
      }
    }
    __syncthreads();
    // Pass B: opened = dilate(nxt); s |= cur & ~opened  (s in global, exclusive words)
    {
      unsigned dp = (r0 == 0) ? 0u : hD(nxt, r0 - 1, c);
      unsigned dc = hD(nxt, r0, c);
#pragma unroll 4
      for (int k = 0; k < 32; ++k) {
        const int r = r0 + k;
        const unsigned dn = (r == H_ - 1) ? 0u : hD(nxt, r + 1, c);
        const unsigned opened = dp | dc | dn;
        const unsigned v = cur[(r << 5) + c] & ~opened;
        if (it == 0) gs[(r << 5) + c] = v;        // first iter overwrites poison
        else         gs[(r << 5) + c] |= v;
        dp = dc; dc = dn;
      }
    }
    __syncthreads();                               // before next pass overwrites cur
  }
}

// ---------------------------------------------------------------------------
// Kernel 2b: word-wise 3x3 dilation of the back skeleton (bsk -> bskd).
// 524288 words total, all L2-resident: negligible cost, and it removes every
// neighborhood load / boundary branch from the bandwidth-critical loss kernel.
// ---------------------------------------------------------------------------
__global__ __launch_bounds__(256) void dilate_kernel(const unsigned* __restrict__ bsk,
                                                     unsigned* __restrict__ bskd) {
  const int w  = blockIdx.x * 256 + threadIdx.x;
  const int iw = w & (IMG_WORDS - 1);
  const int r  = iw >> 5;
  const int c  = iw & 31;
  const unsigned* bm = bsk + (w - iw);             // image base
  unsigned bd = hD(bm, r, c);
  if (r > 0)      bd |= hD(bm, r - 1, c);
  if (r < H_ - 1) bd |= hD(bm, r + 1, c);
  bskd[w] = bd;
}

// ---------------------------------------------------------------------------
// Kernel 3: fused weight-map + CE + partial reduction through WMMA.
// block = 256 threads (8 waves); each wave processes 4 bit-words (128 pixels).
// One block = 1024 pixels = always inside one image -> class weights hoisted.
// Per pair of words the 64 lane values are summed by v_wmma_f32_16x16x4_f32
// with B = ones into an f32 accumulator tile; total = column 0 of C.
// ---------------------------------------------------------------------------
__global__ __launch_bounds__(256) void loss_kernel(const float* __restrict__ pred,
                                                   const unsigned* __restrict__ amask,
                                                   const unsigned* __restrict__ fsk,
                                                   const unsigned* __restrict__ bskd,
                                                   const unsigned* __restrict__ cnts,
                                                   const int* __restrict__ epochp,
                                                   float* __restrict__ partials) {
  __shared__ float red[256];
  const int tid  = threadIdx.x;
  const int lane = tid & 31;
  const int wv   = tid >> 5;
  const float factor = fminf((float)epochp[0] * (1.0f / 50.0f), 1.0f);

  // Per-block (per-image) constants: two divisions per block total.
  const int   img = blockIdx.x >> 10;              // 1024 blocks per image
  const float c1f = (float)cnts[img];
  const float c0f = (float)IMG_PIX - c1f;
  const float mn  = fminf(c0f, c1f);
  const float cw0 = c0f / mn;                      // cw_map for pixels with t==1
  const float cw1 = c1f / mn;                      // cw_map for pixels with t==0
  const float* pbase = pred + (size_t)img * (size_t)(2 * IMG_PIX);

  v8f acc = {};
  float pair0 = 0.0f;
  const int wbase = blockIdx.x * 32 + wv * 4;      // global word index base

#pragma unroll
  for (int kw = 0; kw < 4; ++kw) {
    const int word = wbase + kw;
    const unsigned aw = amask[word];
    const unsigned fw = fsk[word];
    const unsigned bw = bskd[word];                // pre-dilated back skeleton

    const int   t    = (int)((aw >> lane) & 1u);
    const float wmap = 1.0f + factor * (float)(((fw >> lane) & 1u) + ((bw >> lane) & 1u));
    const float cwm  = t ? cw0 : cw1;

    const int qp = (word & (IMG_WORDS - 1)) * 32 + lane;   // pixel within image
    __builtin_prefetch(pbase + qp + 4096, 0, 3);           // global_prefetch_b8
    const float p0 = pbase[qp];
    const float p1 = pbase[IMG_PIX + qp];
    const float z  = t ? (p0 - p1) : (p1 - p0);            // pred_other - pred_t
    const float ce = fmaxf(z, 0.0f) + log1pf(expf(-fabsf(z)));  // stable softplus

    const float v = wmap * cwm * ce;

    if ((kw & 1) == 0) {
      pair0 = v;
    } else {
      v2f A;     A.x = pair0; A.y = v;             // 16x4 A: 2 f32 per lane
      v2f Bones; Bones.x = 1.0f; Bones.y = 1.0f;   // 4x16 B = ones
      acc = __builtin_amdgcn_wmma_f32_16x16x4_f32(
          /*neg_a=*/false, A, /*neg_b=*/false, Bones,
          /*c_mod=*/(short)0, acc, /*reuse_a=*/false, /*reuse_b=*/false);
    }
  }

  // Column N=0 of C lives on lanes 0 (M=0..7) and 16 (M=8..15).
  float ls = 0.0f;
#pragma unroll
  for (int i = 0; i < 8; ++i) ls += acc[i];
  red[tid] = ls;
  __syncthreads();
  if (tid == 0) {
    float s = 0.0f;
    for (int w = 0; w < 8; ++w) s += red[w * 32] + red[w * 32 + 16];
    partials[blockIdx.x] = s;                      // deterministic per block
  }
}

// ---------------------------------------------------------------------------
// Kernel 4: deterministic final reduction of 16384 block partials -> mean
// ---------------------------------------------------------------------------
__global__ __launch_bounds__(1024) void final_kernel(const float* __restrict__ partials,
                                                     float* __restrict__ out) {
  __shared__ float red[1024];
  const int tid = threadIdx.x;
  float s = 0.0f;
#pragma unroll
  for (int i = 0; i < 16; ++i) s += partials[tid + (i << 10)];
  red[tid] = s;
  __syncthreads();
  for (int off = 512; off > 0; off >>= 1) {
    if (tid < off) red[tid] += red[tid + off];
    __syncthreads();
  }
  if (tid == 0) out[0] = red[0] * (1.0f / (float)TOT_PIX);
}

// ---------------------------------------------------------------------------
// Entry point. ws layout (u32 units):
//   [0, 524288)          amask   (packed target bits)
//   [524288, 1048576)    fore_sk (skeleton of a)
//   [1048576, 1572864)   back_sk (skeleton of 1-a, pre-dilate)
//   [1572864, 2097152)   back_sk dilated
//   [2097152, +16)       per-image '1' counts
//   [+32 ...]            16384 f32 block partial sums
// ---------------------------------------------------------------------------
extern "C" void kernel_launch(void* const* d_in, const int* in_sizes, int n_in,
                              void* d_out, int out_size, void* d_ws, size_t ws_size,
                              hipStream_t stream) {
  (void)in_sizes; (void)n_in; (void)out_size; (void)ws_size;
  const float* pred  = (const float*)d_in[0];
  const int*   tgt   = (const int*)d_in[1];
  const int*   epoch = (const int*)d_in[2];

  unsigned* ws       = (unsigned*)d_ws;
  unsigned* amask    = ws;
  unsigned* fsk      = ws + TOT_WORDS;
  unsigned* bsk      = ws + 2 * TOT_WORDS;
  unsigned* bskd     = ws + 3 * TOT_WORDS;
  unsigned* cnts     = ws + 4 * TOT_WORDS;
  float*    partials = (float*)(ws + 4 * TOT_WORDS + 32);

  init_kernel<<<1, 64, 0, stream>>>(cnts);
  pack_kernel<<<TOT_PIX / 256, 256, 0, stream>>>(tgt, amask, cnts);
  sk_kernel<<<dim3(B_, 2), 1024, 0, stream>>>(amask, fsk, bsk);
  dilate_kernel<<<TOT_WORDS / 256, 256, 0, stream>>>(bsk, bskd);
  loss_kernel<<<TOT_PIX / 1024, 256, 0, stream>>>(pred, amask, fsk, bskd, cnts, epoch, partials);
  final_kernel<<<1, 1024, 0, stream>>>(partials, (float*)d_out);
}